// MHCLayerSuperFused_20555713478901
// MI455X (gfx1250) — compile-verified
//
#include <hip/hip_runtime.h>
#include <math.h>

typedef __attribute__((ext_vector_type(2))) float v2f;
typedef __attribute__((ext_vector_type(8))) float v8f;

#define EPS_F 1e-6f
#define C_DIM 4096
#define N_STR 4
#define PF 8   // load-pipeline depth (register ring buffer)

// One workgroup (256 threads = 8 waves) per batch row b.
// Phase 1: V_WMMA_F32_16X16X4_F32 per 16-column tile:
//   A rows 0-3 = Sinkhorn(P), row 4 = sigmoid(H_pre), rows 8-12 duplicate 0-4
//   => D rows 0-3 = P@x (mix), row 4 = x_agg, and lanes 16-31 carry copies so
//      every lane has useful data (branch-free LDS stores, duplicated x_agg).
// Intermediates (5 x 4096 f32 = 80 KB) live in LDS across the RMS reduction.
// Phase 2: out[m][c] = mixed[m][c] + 2*sigmoid(H_post[m]) * (x_agg[c]/rms)*w[c]
__global__ __launch_bounds__(256) void mhc_fused_kernel(
    const float* __restrict__ x,
    const float* __restrict__ H_pre,
    const float* __restrict__ H_post,
    const float* __restrict__ H_res,
    const float* __restrict__ w,
    float* __restrict__ out)
{
    __shared__ float lds_mix[5 * C_DIM];   // rows 0-3: mixed, row 4: x_agg (80 KB)
    __shared__ float red_s[8];
    __shared__ float inv_s;

    const int tid  = threadIdx.x;
    const int lane = tid & 31;
    const int wave = tid >> 5;
    const int b    = blockIdx.x;

    // ---- uniform tiny math (redundant per thread): Sinkhorn(4x4) + gates ----
    float P[4][4];
#pragma unroll
    for (int m = 0; m < 4; ++m)
#pragma unroll
        for (int n = 0; n < 4; ++n)
            P[m][n] = __expf(H_res[m * 4 + n]);
#pragma unroll
    for (int it = 0; it < 3; ++it) {
#pragma unroll
        for (int m = 0; m < 4; ++m) {
            float r = 1.0f / (P[m][0] + P[m][1] + P[m][2] + P[m][3] + EPS_F);
#pragma unroll
            for (int n = 0; n < 4; ++n) P[m][n] *= r;
        }
#pragma unroll
        for (int n = 0; n < 4; ++n) {
            float r = 1.0f / (P[0][n] + P[1][n] + P[2][n] + P[3][n] + EPS_F);
#pragma unroll
            for (int m = 0; m < 4; ++m) P[m][n] *= r;
        }
    }
    float pre[4], post[4];
#pragma unroll
    for (int k = 0; k < 4; ++k) {
        pre[k]  = 1.0f / (1.0f + __expf(-H_pre[k]));
        post[k] = 2.0f / (1.0f + __expf(-H_post[k]));
    }

    // ---- per-lane A fragment (16x4 f32 layout):
    // lanes 0-15: row M=lane, K=0 (v0) / K=1 (v1); lanes 16-31: row M=lane-16, K=2/3.
    // A rows: 0-3 = P, 4 = pre, 8-11 = P, 12 = pre, rest 0.
    const int Mrow = lane & 15;
    const int klo  = (lane < 16) ? 0 : 2;
    const int mr   = Mrow & 7;
    float a0 = 0.0f, a1 = 0.0f;
    if (mr < 4)       { a0 = P[mr][klo]; a1 = P[mr][klo + 1]; }
    else if (mr == 4) { a0 = pre[klo];   a1 = pre[klo + 1];   }
    v2f afrag; afrag.x = a0; afrag.y = a1;

    // ---- phase 1: pipelined stream of x through WMMA, results parked in LDS ----
    const float* xb    = x + (size_t)b * (N_STR * C_DIM);
    const int    nn    = lane & 15;
    const int    kb    = (lane < 16) ? 0 : 2;  // B (4x16): v0 = row kb, v1 = row kb+1
    const int    cwave = wave * 512;           // contiguous 512-column slab per wave
    const float* p0    = xb + (size_t)kb * C_DIM + cwave + nn;
    const float* p1    = p0 + C_DIM;
    const int    ra    = (lane < 16) ? 0 : 2;
    const int    lbase = cwave + nn;

    v2f buf[PF];
#pragma unroll
    for (int i = 0; i < PF; ++i) {
        buf[i].x = p0[16 * i];
        buf[i].y = p1[16 * i];
    }

    float sumsq = 0.0f;
#pragma unroll
    for (int t = 0; t < 32; ++t) {
        v2f bfrag = buf[t & (PF - 1)];
        v8f acc = {0.f, 0.f, 0.f, 0.f, 0.f, 0.f, 0.f, 0.f};
        v8f d = __builtin_amdgcn_wmma_f32_16x16x4_f32(
            false, afrag, false, bfrag, (short)0, acc, false, false);
        // refill this ring slot early (keeps 2*PF loads in flight)
        if (t + PF < 32) {
            buf[t & (PF - 1)].x = p0[16 * (t + PF)];
            buf[t & (PF - 1)].y = p1[16 * (t + PF)];
        }
        const int coff = lbase + 16 * t;
        // lanes 0-15 hold D rows 0-4; lanes 16-31 hold rows 8-12 (duplicates)
        float va = (lane < 16) ? d[0] : d[2];
        float vb = (lane < 16) ? d[1] : d[3];
        lds_mix[ra * C_DIM + coff]       = va;
        lds_mix[(ra + 1) * C_DIM + coff] = vb;
        // all lanes write x_agg (lane pairs write identical values to same addr)
        lds_mix[4 * C_DIM + coff] = d[4];
        sumsq = fmaf(d[4], d[4], sumsq);   // each c counted twice -> halve later
    }

    // ---- block reduction of sum(x_agg^2) -> 1/rms ----
#pragma unroll
    for (int off = 16; off > 0; off >>= 1)
        sumsq += __shfl_down(sumsq, off, 32);
    if (lane == 0) red_s[wave] = sumsq;
    __syncthreads();
    if (tid == 0) {
        float s = 0.0f;
#pragma unroll
        for (int i = 0; i < 8; ++i) s += red_s[i];
        s *= 0.5f;  // every column was accumulated by both lane halves
        inv_s = 1.0f / sqrtf(s * (1.0f / (float)C_DIM) + EPS_F);
    }
    __syncthreads();
    const float inv = inv_s;

    // ---- phase 2: epilogue + coalesced float4 stores ----
    float* ob = out + (size_t)b * (N_STR * C_DIM);
#pragma unroll
    for (int j = 0; j < 4; ++j) {
        const int c = 4 * tid + 1024 * j;
        float4 xa = *(const float4*)&lds_mix[4 * C_DIM + c];
        float4 wv = *(const float4*)&w[c];
        float4 y;
        y.x = xa.x * inv * wv.x;
        y.y = xa.y * inv * wv.y;
        y.z = xa.z * inv * wv.z;
        y.w = xa.w * inv * wv.w;
#pragma unroll
        for (int m = 0; m < 4; ++m) {
            float4 mx = *(const float4*)&lds_mix[m * C_DIM + c];
            float4 o;
            o.x = fmaf(post[m], y.x, mx.x);
            o.y = fmaf(post[m], y.y, mx.y);
            o.z = fmaf(post[m], y.z, mx.z);
            o.w = fmaf(post[m], y.w, mx.w);
            *(float4*)&ob[(size_t)m * C_DIM + c] = o;
        }
    }
}

extern "C" void kernel_launch(void* const* d_in, const int* in_sizes, int n_in,
                              void* d_out, int out_size, void* d_ws, size_t ws_size,
                              hipStream_t stream) {
    const float* x      = (const float*)d_in[0];
    const float* H_pre  = (const float*)d_in[1];
    const float* H_post = (const float*)d_in[2];
    const float* H_res  = (const float*)d_in[3];
    const float* w      = (const float*)d_in[4];
    float* out          = (float*)d_out;

    const int B = in_sizes[0] / (N_STR * C_DIM);
    mhc_fused_kernel<<<B, 256, 0, stream>>>(x, H_pre, H_post, H_res, w, out);
}